// GCNPolicy_17403207483896
// MI455X (gfx1250) — compile-verified
//
#include <hip/hip_runtime.h>

#define N_NODES 50000
#define N_EDGES 800000
#define HID 128
#define N_GRAPHS 128
#define N_ACT 16

typedef float v2f __attribute__((ext_vector_type(2)));
typedef float v8f __attribute__((ext_vector_type(8)));

// ---------------- elementwise helpers ----------------

__global__ void fill_f32_kernel(float* __restrict__ p, float v, unsigned long long n) {
  unsigned long long i = (unsigned long long)blockIdx.x * blockDim.x + threadIdx.x;
  if (i < n) p[i] = v;
}

__global__ void deg_accum_kernel(const int* __restrict__ col, const float* __restrict__ ew,
                                 float* __restrict__ deg, int E) {
  int i = blockIdx.x * blockDim.x + threadIdx.x;
  if (i < E) atomicAdd(&deg[col[i]], ew[i]);
}

__global__ void dinv_kernel(float* __restrict__ deg, int n) {
  int i = blockIdx.x * blockDim.x + threadIdx.x;
  if (i < n) {
    float d = deg[i];
    deg[i] = d > 0.0f ? 1.0f / sqrtf(d) : 0.0f;
  }
}

__global__ void bias_relu_kernel(float* __restrict__ h, const float* __restrict__ b,
                                 unsigned long long n) {
  unsigned long long i = (unsigned long long)blockIdx.x * blockDim.x + threadIdx.x;
  if (i < n) h[i] = fmaxf(h[i] + b[(int)(i & (HID - 1))], 0.0f);
}

// ---------------- WMMA fp32 GEMM: D[Mx128] = A[Mx128] * B[128x128] ----------------
// grid.x = M/16; 256 threads = 8 waves; wave w handles N-tile w (HID/16 == 8).
// V_WMMA_F32_16X16X4_F32 layouts:
//   A (16x4): lanes 0-15 rows M, {v0,v1}={K,K+1}; lanes 16-31 same rows, {K+2,K+3}
//   B (4x16): lanes 0-15 cols N, {v0,v1}={K,K+1}; lanes 16-31 same cols, {K+2,K+3}
//   C/D: VGPR r -> row r (lanes 0-15) / row r+8 (lanes 16-31), col = lane&15
__global__ void gemm128_wmma_kernel(const float* __restrict__ A, const float* __restrict__ B,
                                    float* __restrict__ D) {
  const int wave = threadIdx.x >> 5;
  const int lane = threadIdx.x & 31;
  const int half = lane >> 4;
  const int lr   = lane & 15;
  const int m0 = blockIdx.x * 16;
  const int n0 = wave * 16;
  const float* __restrict__ arow = A + (size_t)(m0 + lr) * HID + half * 2;
  const float* __restrict__ bcol = B + (size_t)(half * 2) * HID + n0 + lr;
  v8f acc = {};
#pragma unroll 8
  for (int k = 0; k < HID; k += 4) {
    v2f a = *(const v2f*)(arow + k);        // two consecutive K values -> global_load_b64
    v2f b;
    b.x = bcol[(size_t)k * HID];
    b.y = bcol[(size_t)k * HID + HID];
    acc = __builtin_amdgcn_wmma_f32_16x16x4_f32(false, a, false, b, (short)0, acc,
                                                false, false);
  }
  float* __restrict__ drow = D + (size_t)(m0 + half * 8) * HID + n0 + lr;
#pragma unroll
  for (int r = 0; r < 8; ++r) drow[(size_t)r * HID] = acc[r];
}

// ---------------- edge scatter: out[col] += dinv[row]*w*dinv[col] * xw[row] ----------------
// One wave per (edge or self-loop); 32 lanes x float4 = 128 features.
__global__ void aggregate_kernel(const float* __restrict__ xw, const int* __restrict__ row,
                                 const int* __restrict__ col, const float* __restrict__ ew,
                                 const float* __restrict__ dinv, float* __restrict__ out) {
  const int lane = threadIdx.x & 31;
  const int wid = (int)((blockIdx.x * blockDim.x + threadIdx.x) >> 5);
  if (wid >= N_EDGES + N_NODES) return;
  int r, c;
  float w;
  if (wid < N_EDGES) {
    r = row[wid]; c = col[wid]; w = ew[wid];
  } else {
    r = c = wid - N_EDGES; w = 1.0f;          // self loop, weight 1
  }
  const float norm = dinv[r] * w * dinv[c];
  const float4 v = ((const float4*)(xw + (size_t)r * HID))[lane];
  float* __restrict__ dst = out + (size_t)c * HID + lane * 4;
  atomicAdd(dst + 0, norm * v.x);
  atomicAdd(dst + 1, norm * v.y);
  atomicAdd(dst + 2, norm * v.z);
  atomicAdd(dst + 3, norm * v.w);
}

// ---------------- segment max pool (values >= 0 post-ReLU: int-bit max is exact) ----------------
__global__ void pool_max_kernel(const float* __restrict__ h, const int* __restrict__ batch,
                                int* __restrict__ pooled_i) {
  unsigned long long i = (unsigned long long)blockIdx.x * blockDim.x + threadIdx.x;
  if (i < (unsigned long long)N_NODES * HID) {
    int node = (int)(i >> 7);
    int j = (int)(i & 127);
    atomicMax(&pooled_i[batch[node] * HID + j], __float_as_int(h[i]));
  }
}

// ---------------- head: out[128x16] = pooled[128x128] @ Wl[128x16] + bl ----------------
// Single block, 8 waves; wave = M-tile, single N-tile of 16.
__global__ void head_wmma_kernel(const float* __restrict__ P, const float* __restrict__ Wl,
                                 const float* __restrict__ bl, float* __restrict__ out) {
  const int wave = threadIdx.x >> 5;
  const int lane = threadIdx.x & 31;
  const int half = lane >> 4;
  const int lr   = lane & 15;
  const int m0 = wave * 16;
  const float* __restrict__ arow = P + (size_t)(m0 + lr) * HID + half * 2;
  const float* __restrict__ bcol = Wl + (size_t)(half * 2) * N_ACT + lr;
  v8f acc = {};
#pragma unroll 8
  for (int k = 0; k < HID; k += 4) {
    v2f a = *(const v2f*)(arow + k);
    v2f b;
    b.x = bcol[(size_t)k * N_ACT];
    b.y = bcol[(size_t)k * N_ACT + N_ACT];
    acc = __builtin_amdgcn_wmma_f32_16x16x4_f32(false, a, false, b, (short)0, acc,
                                                false, false);
  }
  const float bias = bl[lr];
  float* __restrict__ orow = out + (size_t)(m0 + half * 8) * N_ACT + lr;
#pragma unroll
  for (int r = 0; r < 8; ++r) orow[(size_t)r * N_ACT] = acc[r] + bias;
}

// ---------------- launcher ----------------

extern "C" void kernel_launch(void* const* d_in, const int* in_sizes, int n_in,
                              void* d_out, int out_size, void* d_ws, size_t ws_size,
                              hipStream_t stream) {
  const float* x     = (const float*)d_in[0];
  const int*   ei    = (const int*)d_in[1];   // [2, E]: row then col
  const float* ew    = (const float*)d_in[2];
  const int*   batch = (const int*)d_in[3];
  const float* W1    = (const float*)d_in[4];
  const float* b1    = (const float*)d_in[5];
  const float* W2    = (const float*)d_in[6];
  const float* b2    = (const float*)d_in[7];
  const float* Wl    = (const float*)d_in[8];
  const float* bl    = (const float*)d_in[9];
  float* out = (float*)d_out;

  const int* row = ei;
  const int* col = ei + N_EDGES;

  // workspace carve-up (floats): deg/dinv | bufB (xw / hw2) | bufC (h1 / h2) | pooled
  float* ws     = (float*)d_ws;
  float* deg    = ws;                                    // 50000 (pad to 50176)
  float* bufB   = ws + 50176;                            // 50000*128
  float* bufC   = bufB + (size_t)N_NODES * HID;          // 50000*128
  float* pooled = bufC + (size_t)N_NODES * HID;          // 128*128

  const unsigned long long nh = (unsigned long long)N_NODES * HID;   // 6.4M
  const int blkNH   = (int)((nh + 255) / 256);
  const int blkN    = (N_NODES + 255) / 256;
  const int blkE    = (N_EDGES + 255) / 256;
  const int blkAgg  = (N_EDGES + N_NODES) / 8;           // 1 wave per edge/loop, 8 waves/block

  // degree -> dinv (self-loop contributes 1.0)
  fill_f32_kernel<<<blkN, 256, 0, stream>>>(deg, 1.0f, N_NODES);
  deg_accum_kernel<<<blkE, 256, 0, stream>>>(col, ew, deg, N_EDGES);
  dinv_kernel<<<blkN, 256, 0, stream>>>(deg, N_NODES);

  // ----- layer 1 -----
  gemm128_wmma_kernel<<<N_NODES / 16, 256, 0, stream>>>(x, W1, bufB);
  fill_f32_kernel<<<blkNH, 256, 0, stream>>>(bufC, 0.0f, nh);
  aggregate_kernel<<<blkAgg, 256, 0, stream>>>(bufB, row, col, ew, deg, bufC);
  bias_relu_kernel<<<blkNH, 256, 0, stream>>>(bufC, b1, nh);

  // ----- layer 2 -----
  gemm128_wmma_kernel<<<N_NODES / 16, 256, 0, stream>>>(bufC, W2, bufB);
  fill_f32_kernel<<<blkNH, 256, 0, stream>>>(bufC, 0.0f, nh);
  aggregate_kernel<<<blkAgg, 256, 0, stream>>>(bufB, row, col, ew, deg, bufC);
  bias_relu_kernel<<<blkNH, 256, 0, stream>>>(bufC, b2, nh);

  // ----- global max pool -----
  fill_f32_kernel<<<(N_GRAPHS * HID + 255) / 256, 256, 0, stream>>>(
      pooled, -INFINITY, (unsigned long long)N_GRAPHS * HID);
  pool_max_kernel<<<blkNH, 256, 0, stream>>>(bufC, batch, (int*)pooled);

  // ----- head -----
  head_wmma_kernel<<<1, 256, 0, stream>>>(pooled, Wl, bl, out);
}